// SICConv2d_84550726189077
// MI455X (gfx1250) — compile-verified
//
#include <hip/hip_runtime.h>

typedef __attribute__((ext_vector_type(2))) float v2f;
typedef __attribute__((ext_vector_type(8))) float v8f;

#define O_CH   128
#define G_CNT  4
#define CS_SZ  64
#define SEG_K  576        // C*KH*KW = 64*9
#define HW_PIX 1024       // 32*32
#define N_TOT  4096       // B*HW
#define KB     32         // K-block staged in LDS
#define LDA    36         // padded LDS stride (bank-conflict-free, 16B-aligned rows)

// ---------------------------------------------------------------------------
// Kernel 1: collapse (means, col_idx, dest) into dense weights W[O][SEG].
// dest[e] == e / CS  =>  o = e >> 8, g = (e >> 6) & 3  (G*CS = 256).
// One block per output channel o; scatter via LDS float atomics.
// ---------------------------------------------------------------------------
__global__ __launch_bounds__(256) void build_w_kernel(
    const float* __restrict__ means,
    const int*   __restrict__ col_idx,
    float*       __restrict__ Wm)
{
  __shared__ float row[SEG_K];
  const int o = blockIdx.x;
  const int t = threadIdx.x;

  for (int s = t; s < SEG_K; s += 256) row[s] = 0.0f;
  __syncthreads();

  // e = o*256 + t ; g = t/64
  const int   g   = t >> 6;
  const float val = means[o * G_CNT + g];
  const int   s   = col_idx[o * (G_CNT * CS_SZ) + t];
  atomicAdd(&row[s], val);
  __syncthreads();

  for (int s2 = t; s2 < SEG_K; s2 += 256) Wm[o * SEG_K + s2] = row[s2];
}

// ---------------------------------------------------------------------------
// Kernel 2: y[b,o,hw] = bias[o] + sum_s W[o,s] * im2col(x)[s, b*HW+hw]
// GEMM M=128, N=4096, K=576 via V_WMMA_F32_16X16X4_F32 (fp32 MACs).
// Block: 256 threads / 8 waves; tile M=128 x N=128; K staged 32 at a time.
// ---------------------------------------------------------------------------
__global__ __launch_bounds__(256) void conv_wmma_kernel(
    const float* __restrict__ x,
    const float* __restrict__ Wm,
    const float* __restrict__ bias,
    float*       __restrict__ out)
{
  __shared__ float As[O_CH * LDA];   // A tile: [m 0..127][k 0..31] (stride 36)
  __shared__ float Bt[128 * LDA];    // B tile transposed: [n 0..127][k 0..31]

  const int t    = threadIdx.x;
  const int lane = t & 31;
  const int wave = t >> 5;
  const int nb   = blockIdx.x * 128;   // base output column of this block
  const int col  = lane & 15;          // N (or M) index within fragment
  const int half = lane >> 4;          // selects K pair {0,1} vs {2,3}

  v8f acc[8];
#pragma unroll
  for (int i = 0; i < 8; ++i) acc[i] = v8f{0,0,0,0,0,0,0,0};

#pragma unroll 1
  for (int k0 = 0; k0 < SEG_K; k0 += KB) {
    // ---- stage A: W[0..127][k0..k0+31], float4 loads, 4 per thread ----
    {
      int i4 = t;                        // 1024 float4 total
#pragma unroll
      for (int j = 0; j < 4; ++j, i4 += 256) {
        const int m  = i4 >> 3;          // 8 float4 per row
        const int kq = (i4 & 7) << 2;
        const float4 v = *(const float4*)(Wm + m * SEG_K + k0 + kq);
        *(float4*)(&As[m * LDA + kq]) = v;
      }
    }
    // ---- stage B: implicit im2col, transposed into Bt[n][k], 16/thread ----
    {
      int i = t;                         // 32*128 = 4096 elements
#pragma unroll
      for (int j = 0; j < 16; ++j, i += 256) {
        const int k  = i >> 7;           // n fastest -> coalesced x reads
        const int nl = i & 127;
        const int s  = k0 + k;           // s = c*9 + kh*3 + kw
        const int c  = s / 9;
        const int kk = s - c * 9;
        const int kh = kk / 3;
        const int kw = kk - kh * 3;
        const int n  = nb + nl;
        const int b  = n >> 10;
        const int hw = n & 1023;
        const int yy = (hw >> 5) + kh - 1;
        const int xx = (hw & 31) + kw - 1;
        float v = 0.0f;
        if ((unsigned)yy < 32u && (unsigned)xx < 32u)
          v = x[((b * 64 + c) << 10) + (yy << 5) + xx];
        Bt[nl * LDA + k] = v;
      }
    }
    __syncthreads();

    // ---- compute: 8 K-steps of 4, 8 M-tiles per wave ----
#pragma unroll
    for (int ks = 0; ks < KB; ks += 4) {
      // B fragment: lane holds B[ks+2h][n], B[ks+2h+1][n] (VGPR0, VGPR1)
      const v2f bf = *(const v2f*)(&Bt[(wave * 16 + col) * LDA + ks + 2 * half]);
#pragma unroll
      for (int mt = 0; mt < 8; ++mt) {
        const v2f af = *(const v2f*)(&As[(mt * 16 + col) * LDA + ks + 2 * half]);
        acc[mt] = __builtin_amdgcn_wmma_f32_16x16x4_f32(
            /*neg_a=*/false, af, /*neg_b=*/false, bf,
            /*c_mod=*/(short)0, acc[mt],
            /*reuse_a=*/false, /*reuse_b=*/false);
      }
    }
    __syncthreads();
  }

  // ---- epilogue: C layout -> row M = mt*16 + v + 8*half, col N = n ----
  const int n   = nb + wave * 16 + col;
  const int b   = n >> 10;
  const int hw  = n & 1023;
  float* outp   = out + b * (O_CH * HW_PIX) + hw;
#pragma unroll
  for (int mt = 0; mt < 8; ++mt) {
#pragma unroll
    for (int v = 0; v < 8; ++v) {
      const int o = mt * 16 + v + 8 * half;
      outp[o * HW_PIX] = acc[mt][v] + bias[o];
    }
  }
}

// ---------------------------------------------------------------------------
extern "C" void kernel_launch(void* const* d_in, const int* in_sizes, int n_in,
                              void* d_out, int out_size, void* d_ws, size_t ws_size,
                              hipStream_t stream)
{
  const float* x       = (const float*)d_in[0];   // (B,C,H,W) f32
  const float* means   = (const float*)d_in[1];   // (O,G)     f32
  const float* bias    = (const float*)d_in[2];   // (O,)      f32
  const int*   col_idx = (const int*)  d_in[3];   // (E,)      i32
  // d_in[4] (dest) is analytically e/CS -> recomputed, not read.

  float* Wm = (float*)d_ws;                       // 128*576*4 = 294,912 B

  build_w_kernel<<<O_CH, 256, 0, stream>>>(means, col_idx, Wm);
  conv_wmma_kernel<<<N_TOT / 128, 256, 0, stream>>>(x, Wm, bias, (float*)d_out);
}